// MPNN_72301479461294
// MI455X (gfx1250) — compile-verified
//
#include <hip/hip_runtime.h>

#define NSAMP 256
#define NPART 128
#define IB    16      // i-rows per workgroup
#define NWAVE 8
#define BOXL  10.0f
#define PI_F  3.14159265358979323846f

typedef float v2f __attribute__((ext_vector_type(2)));
typedef float v8f __attribute__((ext_vector_type(8)));

struct KParams {
  const float* ri;
  const float* hi;
  const float* hij;
  const float* phiW0[3]; const float* phiW1[3]; const float* phib0[3]; const float* phib1[3];
  const float* fW0[3];   const float* fW1[3];   const float* fb0[3];   const float* fb1[3];
  const float* gW0[3];   const float* gW1[3];   const float* gb0[3];   const float* gb1[3];
};

// D = A(16x4 f32) * B(4x16 f32) + C(16x16 f32)
static __device__ inline v8f wmma4(v2f a, v2f b, v8f c) {
  return __builtin_amdgcn_wmma_f32_16x16x4_f32(false, a, false, b, (short)0, c,
                                               false, false);
}

// Branch-free tanh: prefer CDNA5's V_TANH_F32 TRANS op.
static __device__ inline float fast_tanh(float x) {
#if __has_builtin(__builtin_amdgcn_tanhf)
  return __builtin_amdgcn_tanhf(x);
#elif __has_builtin(__builtin_amdgcn_tanh_f32)
  return __builtin_amdgcn_tanh_f32(x);
#else
  // tanh(x) = 2/(1+e^{-2x}) - 1 ; saturates correctly for large |x|
  float e = __expf(-2.0f * x);
#if __has_builtin(__builtin_amdgcn_rcpf)
  float r = __builtin_amdgcn_rcpf(1.0f + e);
#else
  float r = 1.0f / (1.0f + e);
#endif
  return 2.0f * r - 1.0f;
#endif
}

static __device__ inline float gelu_tanh(float x) {
  float x3 = x * x * x;
  return 0.5f * x * (1.0f + fast_tanh(0.7978845608028654f * (x + 0.044715f * x3)));
}

// B-fragment for weight W (din x 8, row-major), K-chunk c.
// B layout (mirror of documented A layout): n = lane%16, K pair {4c+2*(lane/16), +1}.
static __device__ inline v2f wfragB(const float* W, int din, int c, int lo, int hi) {
  int k0 = 4 * c + 2 * hi;
  v2f r;
  r.x = (lo < 8 && k0     < din) ? W[k0 * 8 + lo]       : 0.0f;
  r.y = (lo < 8 && (k0+1) < din) ? W[(k0 + 1) * 8 + lo] : 0.0f;
  return r;
}

static __device__ inline float bfrag(const float* b, int lo) {
  return (lo < 8) ? b[lo] : 0.0f;
}

__global__ __launch_bounds__(256)
void mpnn_kernel(KParams P, float* __restrict__ out) {
  // Edge feature table: [i_local][j][12] = rij0, rij1, normij, (hij|fij)x8, pad0
  __shared__ float Xe[IB * NPART * 12];          // 98304 B
  __shared__ float XI[IB][20];                   // node feats (12) + nusum (8)
  __shared__ float Sw[NWAVE][16][20];            // per-wave activation scratch

  const int s   = blockIdx.x >> 3;
  const int ib  = blockIdx.x & 7;
  const int tid = threadIdx.x;
  const int w    = tid >> 5;
  const int lane = tid & 31;
  const int lo   = lane & 15;      // A row / B,C column index
  const int hi   = lane >> 4;      // K-half / M-half selector

  const float* ri = P.ri + s * NPART * 2;

  // ---- node features: xi = [sin(w r), cos(w r), hi(8)], nusum slots zeroed ----
  if (tid < IB) {
    const float wf = 2.0f * PI_F / BOXL;
    float rx = ri[(ib * IB + tid) * 2 + 0];
    float ry = ri[(ib * IB + tid) * 2 + 1];
    XI[tid][0] = sinf(wf * rx);
    XI[tid][1] = sinf(wf * ry);
    XI[tid][2] = cosf(wf * rx);
    XI[tid][3] = cosf(wf * ry);
    #pragma unroll
    for (int k = 0; k < 8; ++k) XI[tid][4 + k]  = P.hi[k];
    #pragma unroll
    for (int k = 0; k < 8; ++k) XI[tid][12 + k] = 0.0f;
  }

  // ---- edge features: wave w owns i_local {2w, 2w+1}, all j ----
  float h8[8];
  #pragma unroll
  for (int k = 0; k < 8; ++k) h8[k] = P.hij[k];
  for (int it = 0; it < 8; ++it) {
    int e  = it * 32 + lane;                 // 0..255 within wave
    int il = 2 * w + (e >> 7);
    int j  = e & 127;
    int ig = ib * IB + il;
    float dx = ri[ig * 2 + 0] - ri[j * 2 + 0];
    float dy = ri[ig * 2 + 1] - ri[j * 2 + 1];
    float rjx = dx - BOXL * rintf(dx * (1.0f / BOXL));
    float rjy = dy - BOXL * rintf(dy * (1.0f / BOXL));
    float sx = sinf(PI_F * dx * (1.0f / BOXL));
    float sy = sinf(PI_F * dy * (1.0f / BOXL));
    float nrm = (ig == j) ? 0.0f : (sx * sx + sy * sy);
    float* row = &Xe[(il * NPART + j) * 12];
    row[0] = rjx; row[1] = rjy; row[2] = nrm;
    #pragma unroll
    for (int k = 0; k < 8; ++k) row[3 + k] = h8[k];
    row[11] = 0.0f;
  }
  if (lane < 16) Sw[w][lane][19] = 0.0f;     // permanent zero pad (K=19 slot)
  __syncthreads();

  float* Srow = &Sw[w][lo][0];

  for (int g = 0; g < 3; ++g) {
    const bool last = (g == 2);

    // Weight fragments live in registers across all 16 tiles of this wave.
    v2f wp0[3], wp1[2], wf0c[5], wf1c[2];
    #pragma unroll
    for (int c = 0; c < 3; ++c) wp0[c] = wfragB(P.phiW0[g], 11, c, lo, hi);
    #pragma unroll
    for (int c = 0; c < 2; ++c) wp1[c] = wfragB(P.phiW1[g],  8, c, lo, hi);
    float bp0 = bfrag(P.phib0[g], lo);
    float bp1 = bfrag(P.phib1[g], lo);
    float bf0 = 0.0f, bf1 = 0.0f;
    if (!last) {
      #pragma unroll
      for (int c = 0; c < 5; ++c) wf0c[c] = wfragB(P.fW0[g], 19, c, lo, hi);
      #pragma unroll
      for (int c = 0; c < 2; ++c) wf1c[c] = wfragB(P.fW1[g],  8, c, lo, hi);
      bf0 = bfrag(P.fb0[g], lo);
      bf1 = bfrag(P.fb1[g], lo);
    }

    for (int io = 0; io < 2; ++io) {
      int il = 2 * w + io;
      float acc = 0.0f;                       // nusum partial: feature lo, 8 edges
      for (int jb = 0; jb < 8; ++jb) {
        const float* Xrow = &Xe[(il * NPART + jb * 16 + lo) * 12];

        // phi layer 1: K = 11 (pad 12)
        v8f c1;
        #pragma unroll
        for (int v = 0; v < 8; ++v) c1[v] = bp0;
        #pragma unroll
        for (int c = 0; c < 3; ++c) {
          v2f a = *(const v2f*)(Xrow + 4 * c + 2 * hi);
          c1 = wmma4(a, wp0[c], c1);
        }
        // gelu; stash H rows (edge = v+8*hi, col = lo) for re-entry in A layout
        if (lo < 8) {
          #pragma unroll
          for (int v = 0; v < 8; ++v) Sw[w][v + 8 * hi][lo] = gelu_tanh(c1[v]);
        }
        // phi layer 2: K = 8 -> nuij
        v8f c2;
        #pragma unroll
        for (int v = 0; v < 8; ++v) c2[v] = bp1;
        #pragma unroll
        for (int c = 0; c < 2; ++c) {
          v2f a = *(const v2f*)(Srow + 4 * c + 2 * hi);
          c2 = wmma4(a, wp1[c], c2);
        }
        #pragma unroll
        for (int v = 0; v < 8; ++v) acc += c2[v];

        if (!last) {
          // stash nu at S cols 11..18 (col 19 stays 0)
          if (lo < 8) {
            #pragma unroll
            for (int v = 0; v < 8; ++v) Sw[w][v + 8 * hi][11 + lo] = c2[v];
          }
          // f layer 1: K = 19 (pad 20); k<11 from Xe, k>=11 from S
          v8f c3;
          #pragma unroll
          for (int v = 0; v < 8; ++v) c3[v] = bf0;
          // chunks 0,1: pure Xe
          #pragma unroll
          for (int c = 0; c < 2; ++c) {
            v2f a = *(const v2f*)(Xrow + 4 * c + 2 * hi);
            c3 = wmma4(a, wf0c[c], c3);
          }
          // chunk 2: k = {8,9} (Xe) / {10,11} (Xe, S)
          {
            int k0 = 8 + 2 * hi;
            v2f a;
            a.x = Xrow[k0];                       // k=8 or 10, always < 11
            a.y = hi ? Srow[11] : Xrow[9];        // k=11 -> nu[0], k=9 -> Xe
            c3 = wmma4(a, wf0c[2], c3);
          }
          // chunks 3,4: pure S (nu cols 12..18, col 19 = 0)
          #pragma unroll
          for (int c = 3; c < 5; ++c) {
            v2f a = *(const v2f*)(Srow + 4 * c + 2 * hi);
            c3 = wmma4(a, wf0c[c], c3);
          }
          if (lo < 8) {
            #pragma unroll
            for (int v = 0; v < 8; ++v) Sw[w][v + 8 * hi][lo] = gelu_tanh(c3[v]);
          }
          // f layer 2: K = 8 -> fij; write back into edge table cols 3..10
          v8f c4;
          #pragma unroll
          for (int v = 0; v < 8; ++v) c4[v] = bf1;
          #pragma unroll
          for (int c = 0; c < 2; ++c) {
            v2f a = *(const v2f*)(Srow + 4 * c + 2 * hi);
            c4 = wmma4(a, wf1c[c], c4);
          }
          if (lo < 8) {
            #pragma unroll
            for (int v = 0; v < 8; ++v)
              Xe[(il * NPART + jb * 16 + v + 8 * hi) * 12 + 3 + lo] = c4[v];
          }
        }
      }
      // combine M-halves and publish sum_j nuij for node il
      float tot = acc + __shfl_xor(acc, 16, 32);
      if (lane < 8) XI[il][12 + lane] = tot;
    }
    __syncthreads();

    // g MLP: 16 nodes batched in one WMMA tile (wave 0)
    if (w == 0) {
      v2f wg0[5], wg1[2];
      #pragma unroll
      for (int c = 0; c < 5; ++c) wg0[c] = wfragB(P.gW0[g], 20, c, lo, hi);
      #pragma unroll
      for (int c = 0; c < 2; ++c) wg1[c] = wfragB(P.gW1[g],  8, c, lo, hi);
      float bg0 = bfrag(P.gb0[g], lo);
      float bg1 = bfrag(P.gb1[g], lo);
      v8f c1;
      #pragma unroll
      for (int v = 0; v < 8; ++v) c1[v] = bg0;
      #pragma unroll
      for (int c = 0; c < 5; ++c) {
        v2f a = *(const v2f*)(&XI[lo][4 * c + 2 * hi]);
        c1 = wmma4(a, wg0[c], c1);
      }
      if (lo < 8) {
        #pragma unroll
        for (int v = 0; v < 8; ++v) Sw[0][v + 8 * hi][lo] = gelu_tanh(c1[v]);
      }
      v8f c2;
      #pragma unroll
      for (int v = 0; v < 8; ++v) c2[v] = bg1;
      #pragma unroll
      for (int c = 0; c < 2; ++c) {
        v2f a = *(const v2f*)(&Sw[0][lo][4 * c + 2 * hi]);
        c2 = wmma4(a, wg1[c], c2);
      }
      if (lo < 8) {
        #pragma unroll
        for (int v = 0; v < 8; ++v) XI[v + 8 * hi][4 + lo] = c2[v];
      }
    }
    __syncthreads();
  }

  // final xi = [ri_e(4), gi(8)] -> out[s][i][0..11]
  if (tid < IB * 12) {
    int il = tid / 12, c = tid % 12;
    out[((s * NPART) + ib * IB + il) * 12 + c] = XI[il][c];
  }
}

extern "C" void kernel_launch(void* const* d_in, const int* in_sizes, int n_in,
                              void* d_out, int out_size, void* d_ws, size_t ws_size,
                              hipStream_t stream) {
  (void)in_sizes; (void)n_in; (void)out_size; (void)d_ws; (void)ws_size;
  KParams P;
  P.ri  = (const float*)d_in[0];
  P.hi  = (const float*)d_in[1];
  P.hij = (const float*)d_in[2];
  // jax pytree order (sorted dict keys): per layer f{W0,W1,b0,b1}, g{...}, phi{...}
  for (int l = 0; l < 3; ++l) {
    int b = 3 + 12 * l;
    P.fW0[l]   = (const float*)d_in[b + 0];
    P.fW1[l]   = (const float*)d_in[b + 1];
    P.fb0[l]   = (const float*)d_in[b + 2];
    P.fb1[l]   = (const float*)d_in[b + 3];
    P.gW0[l]   = (const float*)d_in[b + 4];
    P.gW1[l]   = (const float*)d_in[b + 5];
    P.gb0[l]   = (const float*)d_in[b + 6];
    P.gb1[l]   = (const float*)d_in[b + 7];
    P.phiW0[l] = (const float*)d_in[b + 8];
    P.phiW1[l] = (const float*)d_in[b + 9];
    P.phib0[l] = (const float*)d_in[b + 10];
    P.phib1[l] = (const float*)d_in[b + 11];
  }
  dim3 grid(NSAMP * 8), block(256);
  mpnn_kernel<<<grid, block, 0, stream>>>(P, (float*)d_out);
}